// PermuteWeightSharing_56023553409202
// MI455X (gfx1250) — compile-verified
//
#include <hip/hip_runtime.h>

// ---------------------------------------------------------------------------
// PermuteWeightSharing on MI455X (gfx1250, wave32, WMMA)
//
// out_tile(16x16) = (Ppos - Pneg)(16x16) @ x_tile(16x16), per 1KiB f32 tile.
// Implemented with 4x chained V_WMMA_F32_16X16X4_F32 (K=16 total).
// M entries are in {-1,0,1} so the WMMA result is bit-exact vs. the
// reference gather/subtract. Kernel is HBM-bandwidth bound (~256 MiB total,
// ~11.5 us floor at 23.3 TB/s); all VMEM ops are 128B-coalesced per wave.
// ---------------------------------------------------------------------------

typedef __attribute__((ext_vector_type(2))) float v2f;
typedef __attribute__((ext_vector_type(8))) float v8f;

#define TILE_ELEMS 256      // 16 slots x 16 cols, f32, contiguous in memory
#define TILES_PER_WAVE 4
#define WAVES_PER_BLOCK 8   // 256 threads = 8 wave32

__global__ __launch_bounds__(WAVES_PER_BLOCK * 32)
void permute_ws_wmma_kernel(const float* __restrict__ x,
                            const float* __restrict__ Ppos,
                            const float* __restrict__ Pneg,
                            float* __restrict__ out,
                            int numTiles)
{
    const int lane        = threadIdx.x & 31;
    const int waveInBlock = threadIdx.x >> 5;
    const int globalWave  = blockIdx.x * WAVES_PER_BLOCK + waveInBlock;

    const int half = lane >> 4;     // 0: lanes 0-15, 1: lanes 16-31
    const int l15  = lane & 15;

    // ---- A fragments: M = Ppos - Pneg (16x16), split into four 16x4 chunks.
    // 16x4 f32 A layout: lanes 0-15 hold row M=lane  (VGPR0=K0, VGPR1=K1);
    //                    lanes 16-31 hold row M=lane-16 (VGPR0=K2, VGPR1=K3).
    v2f a[4];
    {
        const int row = l15;
#pragma unroll
        for (int kk = 0; kk < 4; ++kk) {
            const int k0 = 4 * kk + 2 * half;  // K index for component .x
            const int k1 = k0 + 1;             // K index for component .y
            a[kk].x = Ppos[row * 16 + k0] - Pneg[row * 16 + k0];
            a[kk].y = Ppos[row * 16 + k1] - Pneg[row * 16 + k1];
        }
    }

    // Per-lane base offset inside a 64-float (4-row) K-chunk of the tile.
    // 4x16 f32 B layout: VGPR0: lanes0-15 row K0, lanes16-31 row K2;
    //                    VGPR1: lanes0-15 row K1, lanes16-31 row K3.
    // Rows 4kk..4kk+3 occupy tile[kk*64 .. kk*64+63] (contiguous).
    const int laneBase = half * 32 + l15;  // 0..15 (lo half) or 32..47 (hi half)

    int tile = globalWave * TILES_PER_WAVE;
#pragma unroll
    for (int t = 0; t < TILES_PER_WAVE; ++t, ++tile) {
        if (tile >= numTiles) return;  // wave-uniform: EXEC stays all-1s

        const float* __restrict__ xt = x   + (size_t)tile * TILE_ELEMS;
        float*       __restrict__ ot = out + (size_t)tile * TILE_ELEMS;

        v8f c = {};  // zero accumulator
#pragma unroll
        for (int kk = 0; kk < 4; ++kk) {
            v2f b;
            b.x = xt[kk * 64 + laneBase];        // rows K0/K2 of this chunk
            b.y = xt[kk * 64 + 16 + laneBase];   // rows K1/K3 of this chunk
            // 8 args: (neg_a, A, neg_b, B, c_mod, C, reuse_a, reuse_b)
            c = __builtin_amdgcn_wmma_f32_16x16x4_f32(
                    false, a[kk], false, b, (short)0, c, false, false);
        }

        // D layout: VGPR v -> (M=v, N=l15) lanes 0-15; (M=v+8, N=l15) lanes 16-31.
        // Memory offset of (slot m, col n) within the tile = m*16 + n.
#pragma unroll
        for (int v = 0; v < 8; ++v) {
            ot[(v + half * 8) * 16 + l15] = c[v];
        }
    }
}

extern "C" void kernel_launch(void* const* d_in, const int* in_sizes, int n_in,
                              void* d_out, int out_size, void* d_ws, size_t ws_size,
                              hipStream_t stream) {
    const float* x    = (const float*)d_in[0];  // (512,16,16,256) f32
    const float* Ppos = (const float*)d_in[1];  // (16,16) f32
    const float* Pneg = (const float*)d_in[2];  // (16,16) f32
    float* out = (float*)d_out;

    const int numTiles    = in_sizes[0] / TILE_ELEMS;            // 131072
    const int wavesNeeded = (numTiles + TILES_PER_WAVE - 1) / TILES_PER_WAVE;
    const int blocks      = (wavesNeeded + WAVES_PER_BLOCK - 1) / WAVES_PER_BLOCK;

    permute_ws_wmma_kernel<<<blocks, WAVES_PER_BLOCK * 32, 0, stream>>>(
        x, Ppos, Pneg, out, numTiles);
}